// F1GRU_55594056679935
// MI455X (gfx1250) — compile-verified
//
#include <hip/hip_runtime.h>
#include <math.h>

// ---------------------------------------------------------------------------
// 3-layer GRU (B=256,S=512,D=64,H=256) + LayerNorm + MLP head for gfx1250.
//  * gx = X @ W_ih^T + b_ih precomputed per layer with fp32 WMMA GEMM
//    (32 rows/block for 2x W reuse).
//  * Recurrent scan: 16 blocks x 16 waves; each block owns 16 batch rows,
//    h resident in LDS; wave owns 16 h-cols x 3 gates (3 WMMA tiles,
//    192 fp32 WMMAs/step); first 96 K-values of Whh fragments pinned in
//    VGPRs across all 512 timesteps to cut L2 traffic and load pressure.
//  * Full fp32 via V_WMMA_F32_16X16X4_F32 (matches fp32 reference over the
//    512-step error-compounding recurrence).
// ---------------------------------------------------------------------------

typedef __attribute__((ext_vector_type(2))) float v2f;
typedef __attribute__((ext_vector_type(8))) float v8f;

#define HID   256
#define G3H   768
#define SEQL  512
#define BATCH 256
#define KPRE  96        // K-prefix of Whh fragments held in registers

__device__ __forceinline__ float sigmoidf_(float x) {
    return 1.0f / (1.0f + __expf(-x));
}

// ---------------------------------------------------------------------------
// gx = X @ W^T + bias.  X: [R, K] row-major, W: [768, K] row-major,
// gx: [R, 768].  Block = 32 rows x 768 cols, 512 threads = 16 waves laid out
// as 8 column-groups x 2 row-groups; each wave computes 2x6 16x16 tiles so
// every B fragment feeds two WMMAs.  K in {64, 256}, kshift = log2(K).
// ---------------------------------------------------------------------------
__global__ __launch_bounds__(512)
void gx_gemm_kernel(const float* __restrict__ X, const float* __restrict__ W,
                    const float* __restrict__ bias, float* __restrict__ GX,
                    int K, int kshift) {
    __shared__ float Xs[32][258];     // padded row stride -> conflict-free

    const int r0  = blockIdx.x * 32;
    const int tid = threadIdx.x;

    // Stage the 32xK X tile into LDS (coalesced).
    const int nelem = 32 << kshift;
    const int kmask = K - 1;
    for (int idx = tid; idx < nelem; idx += 512) {
        const int row = idx >> kshift;
        const int k   = idx & kmask;
        Xs[row][k] = X[(size_t)(r0 + row) * K + k];
    }
    __syncthreads();

    const int wave = tid >> 5;
    const int lane = tid & 31;
    const int ln   = lane & 15;       // col-within-tile / A row-within-tile
    const int kh   = lane >> 4;       // which K-pair this half-wave holds
    const int wc   = wave & 7;        // column group (8)
    const int wr   = wave >> 3;       // row group (2)
    const int n0   = wc * 96;         // 6 tiles of 16 cols per wave
    const int rt   = wr * 16;         // row-tile base within block

    v8f acc[2][6];
#pragma unroll
    for (int m = 0; m < 2; ++m)
#pragma unroll
        for (int j = 0; j < 6; ++j)
#pragma unroll
            for (int i = 0; i < 8; ++i) acc[m][j][i] = 0.0f;

    for (int k0 = 0; k0 < K; k0 += 4) {
        v2f a0 = *(const v2f*)&Xs[rt + ln][k0 + 2 * kh];
        v2f a1 = *(const v2f*)&Xs[rt + 16 - 16 + ln][k0 + 2 * kh]; // placeholder (same as a0 when wr fixed)
        // NOTE: each wave owns ONE row-tile pair: rows rt..rt+15 and the
        // other row-group is handled by the other 8 waves; but to double
        // B reuse we let each wave also process the sibling row tile:
        a1 = *(const v2f*)&Xs[(rt ^ 16) + ln][k0 + 2 * kh];
#pragma unroll
        for (int j = 0; j < 6; ++j) {
            const int n = n0 + j * 16 + ln;
            v2f b = *(const v2f*)&W[(size_t)n * K + k0 + 2 * kh];
            acc[0][j] = __builtin_amdgcn_wmma_f32_16x16x4_f32(
                false, a0, false, b, (short)0, acc[0][j], false, false);
            acc[1][j] = __builtin_amdgcn_wmma_f32_16x16x4_f32(
                false, a1, false, b, (short)0, acc[1][j], false, false);
        }
    }

    // C/D layout: VGPR i holds row i (lanes 0-15) / row i+8 (lanes 16-31).
#pragma unroll
    for (int m = 0; m < 2; ++m) {
        const int rowt  = (m == 0) ? rt : (rt ^ 16);
        const int rbase = r0 + rowt + (kh ? 8 : 0);
#pragma unroll
        for (int j = 0; j < 6; ++j) {
            const int n  = n0 + j * 16 + ln;
            const float bz = bias[n];
#pragma unroll
            for (int i = 0; i < 8; ++i) {
                GX[(size_t)(rbase + i) * G3H + n] = acc[m][j][i] + bz;
            }
        }
    }
}

// ---------------------------------------------------------------------------
// Recurrent scan. Block = 16 batch rows, 512 threads = 16 waves, all 512
// timesteps.  Wave w owns h-columns [w*16, w*16+16) for all 3 gates ->
// 3 WMMA tiles, 192 fp32 WMMAs per step.  Whh fragments for K<KPRE are
// preloaded into VGPRs once and reused for all timesteps.
// ---------------------------------------------------------------------------
__global__ __launch_bounds__(512)
void gru_scan_kernel(const float* __restrict__ GX, const float* __restrict__ Whh,
                     const float* __restrict__ bhh, float* __restrict__ Y) {
    __shared__ float hbuf[16][HID + 2];   // padded row stride 258

    const int tid = threadIdx.x;
    const int b0  = blockIdx.x * 16;

    // h0 = 0
    for (int idx = tid; idx < 16 * (HID + 2); idx += 512)
        ((float*)hbuf)[idx] = 0.0f;
    __syncthreads();

    const int wave = tid >> 5;
    const int lane = tid & 31;
    const int ln   = lane & 15;
    const int kh   = lane >> 4;
    const int c0   = wave * 16;          // h-columns owned by this wave
    const int crow = kh ? 8 : 0;         // C-fragment base row

    // Tile j (gate j): column n = j*HID + c0 + ln
    int   ncol[3];
    float biasv[3];
#pragma unroll
    for (int j = 0; j < 3; ++j) {
        ncol[j]  = j * HID + c0 + ln;
        biasv[j] = bhh[ncol[j]];
    }

    // Pin Whh B-fragments for K = 0..KPRE-1 in registers (3 x 24 x v2f).
    v2f Breg[3][KPRE / 4];
#pragma unroll
    for (int j = 0; j < 3; ++j)
#pragma unroll
        for (int q = 0; q < KPRE / 4; ++q)
            Breg[j][q] = *(const v2f*)&Whh[(size_t)ncol[j] * HID + q * 4 + 2 * kh];

    for (int t = 0; t < SEQL; ++t) {
        v8f acc[3];
#pragma unroll
        for (int j = 0; j < 3; ++j)
#pragma unroll
            for (int i = 0; i < 8; ++i) acc[j][i] = biasv[j];

        // gh = h @ Whh^T : register-resident B for K < KPRE ...
#pragma unroll
        for (int q = 0; q < KPRE / 4; ++q) {
            v2f a = *(const v2f*)&hbuf[ln][q * 4 + 2 * kh];
#pragma unroll
            for (int j = 0; j < 3; ++j)
                acc[j] = __builtin_amdgcn_wmma_f32_16x16x4_f32(
                    false, a, false, Breg[j][q], (short)0, acc[j], false, false);
        }
        // ... streamed B for the rest (L2-resident).
        for (int k0 = KPRE; k0 < HID; k0 += 4) {
            v2f a = *(const v2f*)&hbuf[ln][k0 + 2 * kh];
#pragma unroll
            for (int j = 0; j < 3; ++j) {
                v2f b = *(const v2f*)&Whh[(size_t)ncol[j] * HID + k0 + 2 * kh];
                acc[j] = __builtin_amdgcn_wmma_f32_16x16x4_f32(
                    false, a, false, b, (short)0, acc[j], false, false);
            }
        }

        // Fused gate math: acc[0]=r-side, acc[1]=z-side, acc[2]=n-side.
        const int c = c0 + ln;
        float hnew[8];
#pragma unroll
        for (int i = 0; i < 8; ++i) {
            const int row = crow + i;
            const size_t base = ((size_t)(b0 + row) * SEQL + t) * G3H;
            const float xr = GX[base + c];
            const float xz = GX[base + HID + c];
            const float xn = GX[base + 2 * HID + c];
            const float r  = sigmoidf_(xr + acc[0][i]);
            const float z  = sigmoidf_(xz + acc[1][i]);
            const float nn = tanhf(xn + r * acc[2][i]);
            hnew[i] = (1.0f - z) * nn + z * hbuf[row][c];
        }

        __syncthreads();   // all waves done reading old h
#pragma unroll
        for (int i = 0; i < 8; ++i) {
            const int row = crow + i;
            hbuf[row][c] = hnew[i];
            Y[((size_t)(b0 + row) * SEQL + t) * HID + c] = hnew[i];
        }
        __syncthreads();   // new h visible to everyone
    }
}

// ---------------------------------------------------------------------------
// LayerNorm + MLP head. One thread per batch row (tiny workload).
// ---------------------------------------------------------------------------
__global__ __launch_bounds__(256)
void head_kernel(const float* __restrict__ Ylast, int rowStride,
                 const float* __restrict__ gamma, const float* __restrict__ beta,
                 const float* __restrict__ W1, const float* __restrict__ b1,
                 const float* __restrict__ W2, const float* __restrict__ b2,
                 float* __restrict__ out) {
    const int b = blockIdx.x * blockDim.x + threadIdx.x;
    if (b >= BATCH) return;
    const float* row = Ylast + (size_t)b * rowStride;

    float mu = 0.0f;
    for (int k = 0; k < HID; ++k) mu += row[k];
    mu *= (1.0f / HID);
    float var = 0.0f;
    for (int k = 0; k < HID; ++k) { const float d = row[k] - mu; var += d * d; }
    var *= (1.0f / HID);
    const float rs = rsqrtf(var + 1e-5f);

    float pred = b2[0];
    for (int j = 0; j < 32; ++j) {
        float s = b1[j];
        for (int k = 0; k < HID; ++k) {
            const float xn = (row[k] - mu) * rs * gamma[k] + beta[k];
            s += xn * W1[j * HID + k];
        }
        pred += fmaxf(s, 0.0f) * W2[j];
    }
    out[b] = pred;
}

// ---------------------------------------------------------------------------
extern "C" void kernel_launch(void* const* d_in, const int* in_sizes, int n_in,
                              void* d_out, int out_size, void* d_ws, size_t ws_size,
                              hipStream_t stream) {
    (void)in_sizes; (void)n_in; (void)out_size; (void)ws_size;

    const float* x         = (const float*)d_in[0];   // [256,512,64]
    const float* W_ih0     = (const float*)d_in[1];   // [768,64]
    const float* W_ih_rest = (const float*)d_in[2];   // [2,768,256]
    const float* W_hh      = (const float*)d_in[3];   // [3,768,256]
    const float* b_ih      = (const float*)d_in[4];   // [3,768]
    const float* b_hh      = (const float*)d_in[5];   // [3,768]
    const float* ln_g      = (const float*)d_in[6];   // [256]
    const float* ln_b      = (const float*)d_in[7];   // [256]
    const float* W1        = (const float*)d_in[8];   // [32,256]
    const float* b1        = (const float*)d_in[9];   // [32]
    const float* W2        = (const float*)d_in[10];  // [1,32]
    const float* b2        = (const float*)d_in[11];  // [1]
    float* out = (float*)d_out;

    const size_t R = (size_t)BATCH * SEQL;            // 131072 rows
    float* gx   = (float*)d_ws;                       // R * 768 floats (384 MB)
    float* seqA = gx + R * (size_t)G3H;               // R * 256 floats (128 MB)
    float* seqB = seqA + R * (size_t)HID;             // R * 256 floats (128 MB)

    const dim3 blkG(512);
    const dim3 gGemm((unsigned)(R / 32));
    const dim3 blkS(512);
    const dim3 gScan(BATCH / 16);

    // ---- layer 0 (K = 64) ----
    gx_gemm_kernel<<<gGemm, blkG, 0, stream>>>(x, W_ih0, b_ih, gx, 64, 6);
    gru_scan_kernel<<<gScan, blkS, 0, stream>>>(gx, W_hh, b_hh, seqA);

    // ---- layer 1 (K = 256) ----
    gx_gemm_kernel<<<gGemm, blkG, 0, stream>>>(seqA, W_ih_rest, b_ih + G3H, gx, 256, 8);
    gru_scan_kernel<<<gScan, blkS, 0, stream>>>(gx, W_hh + (size_t)G3H * HID,
                                                b_hh + G3H, seqB);

    // ---- layer 2 (K = 256) ----
    gx_gemm_kernel<<<gGemm, blkG, 0, stream>>>(seqB, W_ih_rest + (size_t)G3H * HID,
                                               b_ih + 2 * G3H, gx, 256, 8);
    gru_scan_kernel<<<gScan, blkS, 0, stream>>>(gx, W_hh + 2 * (size_t)G3H * HID,
                                                b_hh + 2 * G3H, seqA);

    // ---- LayerNorm + MLP head on h[:, S-1, :] ----
    head_kernel<<<dim3(1), dim3(256), 0, stream>>>(seqA + (size_t)(SEQL - 1) * HID,
                                                   SEQL * HID, ln_g, ln_b,
                                                   W1, b1, W2, b2, out);
}